// E3NNModel_1563368095919
// MI455X (gfx1250) — compile-verified
//
#include <hip/hip_runtime.h>
#include <math.h>

// Problem constants (from reference)
#define BATCH   8
#define NNODE   64
#define NODES   512      // BATCH*NNODE
#define NBASIS  20
#define HID     32
#define EPAD    32768    // BATCH*NNODE*NNODE padded edges (diagonal masked later)
#define MTILES  2048     // EPAD/16

// scale = (1/sqrt(64)) * (1/sqrt(32))
#define SCALE   0.022097086912079612f

// Workspace layout (float offsets)
#define OFF_WC     0          // 32*32      w2 contracted with w_read
#define OFF_BC     1024       // 32         b2 contracted with w_read
#define OFF_WSR    1056       // 32         w_self contracted with w_read
#define OFF_G0R    1088       // 512*32     per-node reduced weight
#define OFF_B0R    17472      // 512
#define OFF_SELFR  17984      // 512
#define OFF_RAD    18496      // 32768*20   gaussian radial basis
#define OFF_HID    673856     // 32768*32   relu(radial@w1+b1)
#define OFF_PART   1722432    // 128        per-block partial energy sums
// total = 1722560 floats ~ 6.9 MB

typedef float v2f __attribute__((ext_vector_type(2)));
typedef float v8f __attribute__((ext_vector_type(8)));

// ---------------------------------------------------------------------------
// K1: contract w2/b2/w_self with w_read  ->  wc[32][32], bc[32], wsr[32]
// ---------------------------------------------------------------------------
__global__ void prep_kernel(const float* __restrict__ w2,
                            const float* __restrict__ b2,
                            const float* __restrict__ w_self,
                            const float* __restrict__ w_read,
                            float* __restrict__ ws) {
    __shared__ float wr[64];
    int t = threadIdx.x;
    if (t < 64) wr[t] = w_read[t];
    __syncthreads();

    float* wc  = ws + OFF_WC;
    float* bc  = ws + OFF_BC;
    float* wsr = ws + OFF_WSR;

    for (int idx = t; idx < 1024; idx += 256) {
        int k = idx >> 5, u = idx & 31;
        const float* src = w2 + (size_t)k * 3072 + (size_t)u * 64;
        float s = 0.f;
        for (int w = 0; w < 64; ++w) s += src[w] * wr[w];
        wc[idx] = s;
    }
    if (t < 32) {
        const float* sb = b2 + (size_t)t * 64;
        const float* sw = w_self + (size_t)t * 64;
        float s0 = 0.f, s1 = 0.f;
        for (int w = 0; w < 64; ++w) { s0 += sb[w] * wr[w]; s1 += sw[w] * wr[w]; }
        bc[t]  = s0;
        wsr[t] = s1;
    }
}

// ---------------------------------------------------------------------------
// K2: per-node embed (argmax) + reduced weights g0r[j][32], b0r[j], selfr[j]
//     grid: 512 blocks x 32 threads (one wave per node)
// ---------------------------------------------------------------------------
__global__ void node_kernel(const float* __restrict__ node_attrs,
                            const float* __restrict__ atom_emb_w,
                            float* __restrict__ ws) {
    int j = blockIdx.x;        // node id 0..511
    int u = threadIdx.x;       // 0..31

    float a0 = node_attrs[j * 3 + 0];
    float a1 = node_attrs[j * 3 + 1];
    float a2 = node_attrs[j * 3 + 2];
    int z = 0; float best = a0;
    if (a1 > best) { best = a1; z = 1; }
    if (a2 > best) { z = 2; }

    __shared__ float hrow[32];
    hrow[u] = atom_emb_w[z * 32 + u];
    __syncthreads();

    const float* wc = ws + OFF_WC;
    float s = 0.f;
    for (int q = 0; q < 32; ++q) s += hrow[q] * wc[u * 32 + q];   // k = u
    ws[OFF_G0R + (size_t)j * 32 + u] = s;

    if (u == 0) {
        const float* bc = ws + OFF_BC;
        float sb = 0.f;
        for (int q = 0; q < 32; ++q) sb += hrow[q] * bc[q];
        ws[OFF_B0R + j] = sb;
    }
    if (u == 1) {
        const float* wsr = ws + OFF_WSR;
        float ss = 0.f;
        for (int q = 0; q < 32; ++q) ss += hrow[q] * wsr[q];
        ws[OFF_SELFR + j] = ss;
    }
}

// ---------------------------------------------------------------------------
// K3: radial basis per padded edge: rad[e][p] = exp(-gamma*(|ri-rj|-c_p)^2)
//     grid: 128 blocks x 256 threads = 32768 edges
// ---------------------------------------------------------------------------
__global__ void radial_kernel(const float* __restrict__ pos,
                              const float* __restrict__ centers,
                              const float* __restrict__ gamma,
                              float* __restrict__ ws) {
    int e = blockIdx.x * 256 + threadIdx.x;      // 0..32767
    int b = e >> 12;
    int i = (e >> 6) & 63;
    int j = e & 63;

    const float* pi = pos + ((size_t)(b * 64 + i)) * 3;
    const float* pj = pos + ((size_t)(b * 64 + j)) * 3;
    float dx = pi[0] - pj[0];
    float dy = pi[1] - pj[1];
    float dz = pi[2] - pj[2];
    float len = sqrtf(dx * dx + dy * dy + dz * dz);
    float g = gamma[0];

    float* r = ws + OFF_RAD + (size_t)e * NBASIS;
    for (int p = 0; p < NBASIS; ++p) {
        float d = len - centers[p];
        r[p] = expf(-g * d * d);
    }
}

// ---------------------------------------------------------------------------
// K4: hid = relu(radial @ w1 + b1)  via V_WMMA_F32_16X16X4_F32
//     M=32768 (16/tile), K=20 (5 steps of 4), N=32 (2 tiles per wave)
//     grid: 2048 blocks x 32 threads (one wave per M-tile)
// Fragment layouts per CDNA5 ISA 7.12.2:
//   A 16x4 : lane L (L<16): M=L,    K={k0,k0+1};  L>=16: M=L-16, K={k0+2,k0+3}
//   B 4x16 : lane L (L<16): N=L,    K={k0,k0+1};  L>=16: N=L-16, K={k0+2,k0+3}
//   C 16x16: VGPR r -> M = r + 8*(L/16), N = L%16
// ---------------------------------------------------------------------------
__global__ void hid_gemm_kernel(const float* __restrict__ w1,
                                const float* __restrict__ b1,
                                float* __restrict__ ws) {
    const float* rad = ws + OFF_RAD;
    float*       hid = ws + OFF_HID;

    int lane = threadIdx.x;
    int m  = lane & 15;
    int hi = lane >> 4;                      // half-wave select
    size_t tile = blockIdx.x;                // 0..2047

    const float* Arow = rad + (tile * 16 + m) * NBASIS + 2 * hi;

    v8f c0 = {0.f, 0.f, 0.f, 0.f, 0.f, 0.f, 0.f, 0.f};
    v8f c1 = {0.f, 0.f, 0.f, 0.f, 0.f, 0.f, 0.f, 0.f};

    for (int k0 = 0; k0 < NBASIS; k0 += 4) {
        v2f a = *(const v2f*)(Arow + k0);    // radial[e][k0+2*hi .. +1], 8B aligned
        int kr = k0 + 2 * hi;
        v2f bA, bB;
        bA.x = w1[(kr + 0) * 32 + m];        // N-tile 0: n = m
        bA.y = w1[(kr + 1) * 32 + m];
        bB.x = w1[(kr + 0) * 32 + 16 + m];   // N-tile 1: n = 16 + m
        bB.y = w1[(kr + 1) * 32 + 16 + m];
        c0 = __builtin_amdgcn_wmma_f32_16x16x4_f32(false, a, false, bA,
                                                   (short)0, c0, false, false);
        c1 = __builtin_amdgcn_wmma_f32_16x16x4_f32(false, a, false, bB,
                                                   (short)0, c1, false, false);
    }

    float bias_lo = b1[m];
    float bias_hi = b1[16 + m];
    for (int r = 0; r < 8; ++r) {
        size_t e = tile * 16 + (size_t)(hi * 8 + r);   // row M = hi*8 + r
        hid[e * 32 + m]      = fmaxf(c0[r] + bias_lo, 0.f);
        hid[e * 32 + 16 + m] = fmaxf(c1[r] + bias_hi, 0.f);
    }
}

// ---------------------------------------------------------------------------
// K5: per-edge energy dot:  sum over edges (i!=j) of hid[e,:]·g0r[col,:]
//     grid: 128 blocks x 256 threads; block-local tree reduce -> partial[blk]
// ---------------------------------------------------------------------------
__global__ void edge_energy_kernel(float* __restrict__ ws) {
    int tid = threadIdx.x;
    int e = blockIdx.x * 256 + tid;          // 0..32767, single batch per block
    int b = e >> 12;
    int i = (e >> 6) & 63;
    int j = e & 63;

    float dot = 0.f;
    if (i != j) {
        const float* h = ws + OFF_HID + (size_t)e * 32;
        const float* g = ws + OFF_G0R + (size_t)(b * 64 + j) * 32;
        for (int k = 0; k < 32; ++k) dot += h[k] * g[k];
    }

    __shared__ float red[256];
    red[tid] = dot;
    __syncthreads();
    for (int s = 128; s > 0; s >>= 1) {
        if (tid < s) red[tid] += red[tid + s];
        __syncthreads();
    }
    if (tid == 0) ws[OFF_PART + blockIdx.x] = red[0];
}

// ---------------------------------------------------------------------------
// K6: finalize: out[b] = SCALE * ( Σ_blk partial + 63*Σ_j b0r + Σ_i selfr )
//     grid: 8 blocks x 64 threads
// ---------------------------------------------------------------------------
__global__ void final_kernel(const float* __restrict__ ws,
                             float* __restrict__ out) {
    int b = blockIdx.x;
    int t = threadIdx.x;                     // 0..63

    float val = 63.f * ws[OFF_B0R + b * 64 + t] + ws[OFF_SELFR + b * 64 + t];
    if (t < 16) val += ws[OFF_PART + b * 16 + t];   // 16 edge blocks per batch

    __shared__ float red[64];
    red[t] = val;
    __syncthreads();
    for (int s = 32; s > 0; s >>= 1) {
        if (t < s) red[t] += red[t + s];
        __syncthreads();
    }
    if (t == 0) out[b] = SCALE * red[0];
}

// ---------------------------------------------------------------------------
extern "C" void kernel_launch(void* const* d_in, const int* in_sizes, int n_in,
                              void* d_out, int out_size, void* d_ws, size_t ws_size,
                              hipStream_t stream) {
    const float* pos        = (const float*)d_in[0];   // (8,64,3)
    const float* node_attrs = (const float*)d_in[1];   // (8,64,3)
    const float* atom_emb_w = (const float*)d_in[2];   // (3,32)
    const float* centers    = (const float*)d_in[3];   // (20,)
    const float* gamma      = (const float*)d_in[4];   // (1,)
    const float* w1         = (const float*)d_in[5];   // (20,32)
    const float* b1         = (const float*)d_in[6];   // (32,)
    const float* w2         = (const float*)d_in[7];   // (32,3072)
    const float* b2         = (const float*)d_in[8];   // (3072,)
    const float* w_self     = (const float*)d_in[9];   // (32,64)
    const float* w_read     = (const float*)d_in[10];  // (64,1)
    float* out = (float*)d_out;                        // (8,1)
    float* ws  = (float*)d_ws;

    prep_kernel<<<1, 256, 0, stream>>>(w2, b2, w_self, w_read, ws);
    node_kernel<<<NODES, 32, 0, stream>>>(node_attrs, atom_emb_w, ws);
    radial_kernel<<<EPAD / 256, 256, 0, stream>>>(pos, centers, gamma, ws);
    hid_gemm_kernel<<<MTILES, 32, 0, stream>>>(w1, b1, ws);
    edge_energy_kernel<<<EPAD / 256, 256, 0, stream>>>(ws);
    final_kernel<<<BATCH, 64, 0, stream>>>(ws, out);
}